// TemporalGATLayer_19567871001360
// MI455X (gfx1250) — compile-verified
//
#include <hip/hip_runtime.h>
#include <stdint.h>

#define N_NODES 100000
#define N_EDGES 1600000

typedef float v2f __attribute__((ext_vector_type(2)));
typedef float v8f __attribute__((ext_vector_type(8)));

// ---------- order-preserving float <-> uint key ----------
__device__ __forceinline__ unsigned fkey(float f) {
  unsigned b = __float_as_uint(f);
  return (b & 0x80000000u) ? ~b : (b | 0x80000000u);
}
__device__ __forceinline__ float funkey(unsigned k) {
  return __uint_as_float((k & 0x80000000u) ? (k ^ 0x80000000u) : ~k);
}

// ---------- timestamps min/max ----------
__global__ __launch_bounds__(256) void ts_minmax(const float* __restrict__ ts,
                                                 unsigned* __restrict__ keys, int n) {
  __shared__ unsigned smin[256], smax[256];
  unsigned mn = 0xFFFFFFFFu, mx = 0u;
  for (int i = blockIdx.x * blockDim.x + threadIdx.x; i < n; i += gridDim.x * blockDim.x) {
    unsigned k = fkey(ts[i]);
    mn = min(mn, k); mx = max(mx, k);
  }
  const int t = threadIdx.x;
  smin[t] = mn; smax[t] = mx;
  __syncthreads();
  for (int s = 128; s > 0; s >>= 1) {
    if (t < s) { smin[t] = min(smin[t], smin[t + s]); smax[t] = max(smax[t], smax[t + s]); }
    __syncthreads();
  }
  if (t == 0) { atomicMin(&keys[0], smin[0]); atomicMax(&keys[1], smax[0]); }
}

// ---------- fused QKV projection via f32 WMMA ----------
// Block = 256 threads (8 waves), one 16-row tile of x per block staged in LDS.
// Wave w computes the 16-col block col0=w*16 of q, k and v simultaneously.
__global__ __launch_bounds__(256) void qkv_gemm_wmma(
    const float* __restrict__ x,
    const float* __restrict__ Wq, const float* __restrict__ Wk, const float* __restrict__ Wv,
    float* __restrict__ q, float* __restrict__ k, float* __restrict__ v)
{
  __shared__ float xs[16 * 128];
  const int tid = threadIdx.x;
  const int m0  = blockIdx.x * 16;

  // stage 16x128 f32 tile of x (2048 floats, 2 x float4 per thread)
  const float4* xg  = (const float4*)(x + (size_t)m0 * 128);
  float4*       xs4 = (float4*)xs;
  xs4[tid]       = xg[tid];
  xs4[tid + 256] = xg[tid + 256];
  __syncthreads();

  const int lane  = tid & 31;
  const int wave  = tid >> 5;
  const int col0  = wave * 16;
  const int mrow  = lane & 15;          // M (A) / N (B,C)
  const int khalf = (lane >> 4) << 1;   // 0 for lanes 0-15, 2 for lanes 16-31
  const int colB  = col0 + mrow;

  v8f cq = {}, ck = {}, cv = {};
  #pragma unroll
  for (int k0 = 0; k0 < 128; k0 += 4) {
    // A fragment: lane holds x[m0+mrow][k0+khalf .. +1]
    const float* arow = xs + mrow * 128 + k0 + khalf;
    v2f a; a.x = arow[0]; a.y = arow[1];
    // B fragments: lane holds W[k0+khalf .. +1][colB]
    const int r0 = (k0 + khalf) * 128 + colB;
    v2f bq, bk, bv;
    bq.x = Wq[r0]; bq.y = Wq[r0 + 128];
    bk.x = Wk[r0]; bk.y = Wk[r0 + 128];
    bv.x = Wv[r0]; bv.y = Wv[r0 + 128];
    cq = __builtin_amdgcn_wmma_f32_16x16x4_f32(false, a, false, bq, (short)0, cq, false, false);
    ck = __builtin_amdgcn_wmma_f32_16x16x4_f32(false, a, false, bk, (short)0, ck, false, false);
    cv = __builtin_amdgcn_wmma_f32_16x16x4_f32(false, a, false, bv, (short)0, cv, false, false);
  }

  // C/D layout: VGPR r -> row m0 + r + 8*(lane>=16), col colB
  const int rowHi = (lane >> 4) * 8;
  #pragma unroll
  for (int r = 0; r < 8; ++r) {
    const size_t off = (size_t)(m0 + r + rowHi) * 128 + colB;
    q[off] = cq[r]; k[off] = ck[r]; v[off] = cv[r];
  }
}

// ---------- edge pass 1: attention logits -> exp, segment sums ----------
// lane = (edge, head): gid>>3 = edge, gid&7 = head. 8 lanes of one edge read
// contiguous 512B blocks of q[src]/k[dst] (L2-resident: qkv = 154MB < 192MB L2).
__global__ __launch_bounds__(256) void edge_pass1(
    const int*   __restrict__ ei,
    const float* __restrict__ eattr,
    const float* __restrict__ ts,
    const float* __restrict__ q,
    const float* __restrict__ k,
    const float* __restrict__ freq,
    const float* __restrict__ tbias,
    const float* __restrict__ ew,
    const float* __restrict__ eb,
    const unsigned* __restrict__ keys,
    float* __restrict__ aexp,
    float* __restrict__ asum)
{
  __shared__ float s_ew[69 * 8];
  __shared__ float s_freq[32];
  __shared__ float s_tb[8], s_eb[8];
  const int tid = threadIdx.x;
  for (int i = tid; i < 69 * 8; i += 256) s_ew[i] = ew[i];
  if (tid < 32) s_freq[tid] = freq[tid];
  if (tid < 8) { s_tb[tid] = tbias[tid]; s_eb[tid] = eb[tid]; }
  __syncthreads();

  const size_t gid = (size_t)blockIdx.x * 256 + tid;
  const int e = (int)(gid >> 3);
  const int h = (int)(gid & 7);

  const int src = ei[e];
  const int dst = ei[N_EDGES + e];

  // q[src,h,:] . k[dst,h,:] / sqrt(16)
  const float4* qp = (const float4*)(q + (size_t)src * 128 + h * 16);
  const float4* kp = (const float4*)(k + (size_t)dst * 128 + h * 16);
  float acc = 0.f;
  #pragma unroll
  for (int i = 0; i < 4; ++i) {
    float4 qa = qp[i], ka = kp[i];
    acc += qa.x * ka.x + qa.y * ka.y + qa.z * ka.z + qa.w * ka.w;
  }
  float attn = acc * 0.25f;

  // temporal bias term
  const float t    = ts[e];
  const float tmin = funkey(keys[0]);
  const float tmax = funkey(keys[1]);
  const float t_norm = (tmax > tmin) ? (t - tmin) / (tmax - tmin + 1e-6f) : 1.0f;
  attn += s_tb[h] * t_norm + s_eb[h];

  // edge_attr part of edge_bias
  #pragma unroll
  for (int j = 0; j < 5; ++j) attn += eattr[(size_t)e * 5 + j] * s_ew[j * 8 + h];

  // time-encoding part: [sin f0, cos f0, sin f1, cos f1, ...] @ edge_w[5:,h]
  const float tn = log1pf(t + 1e-6f);
  #pragma unroll
  for (int i = 0; i < 32; ++i) {
    float s, c;
    __sincosf(tn * s_freq[i], &s, &c);
    attn += s * s_ew[(5 + 2 * i) * 8 + h] + c * s_ew[(6 + 2 * i) * 8 + h];
  }

  attn = (attn >= 0.f) ? attn : 0.2f * attn;   // leaky relu
  const float ax = __expf(attn);
  aexp[gid] = ax;
  atomicAdd(&asum[(size_t)dst * 8 + h], ax);
}

// ---------- edge pass 2: normalize + scatter weighted v ----------
__global__ __launch_bounds__(256) void edge_pass2(
    const int*   __restrict__ ei,
    const float* __restrict__ v,
    const float* __restrict__ aexp,
    const float* __restrict__ asum,
    float* __restrict__ out)
{
  const size_t gid = (size_t)blockIdx.x * 256 + threadIdx.x;
  const int e = (int)(gid >> 3);
  const int h = (int)(gid & 7);
  const int src = ei[e];
  const int dst = ei[N_EDGES + e];

  const float w = aexp[gid] / (asum[(size_t)dst * 8 + h] + 1e-8f);
  const float4* vp = (const float4*)(v + (size_t)src * 128 + h * 16);
  float* op = out + (size_t)dst * 128 + h * 16;
  #pragma unroll
  for (int i = 0; i < 4; ++i) {
    float4 vv = vp[i];
    atomicAdd(op + 4 * i + 0, w * vv.x);
    atomicAdd(op + 4 * i + 1, w * vv.y);
    atomicAdd(op + 4 * i + 2, w * vv.z);
    atomicAdd(op + 4 * i + 3, w * vv.w);
  }
}

extern "C" void kernel_launch(void* const* d_in, const int* in_sizes, int n_in,
                              void* d_out, int out_size, void* d_ws, size_t ws_size,
                              hipStream_t stream) {
  (void)in_sizes; (void)n_in; (void)ws_size;
  const float* x     = (const float*)d_in[0];
  const int*   ei    = (const int*)  d_in[1];
  const float* eattr = (const float*)d_in[2];
  const float* ts    = (const float*)d_in[3];
  const float* Wq    = (const float*)d_in[4];
  const float* Wk    = (const float*)d_in[5];
  const float* Wv    = (const float*)d_in[6];
  const float* freq  = (const float*)d_in[7];
  const float* tb    = (const float*)d_in[8];
  const float* ew    = (const float*)d_in[9];
  const float* eb    = (const float*)d_in[10];
  float* out = (float*)d_out;

  // workspace layout (floats): q | k | v | attn_exp | attn_sum | keys(2 u32)
  float* q    = (float*)d_ws;
  float* k    = q    + (size_t)N_NODES * 128;
  float* v    = k    + (size_t)N_NODES * 128;
  float* aexp = v    + (size_t)N_NODES * 128;
  float* asum = aexp + (size_t)N_EDGES * 8;
  unsigned* keys = (unsigned*)(asum + (size_t)N_NODES * 8);

  hipMemsetAsync(out,  0, (size_t)out_size * sizeof(float), stream);
  hipMemsetAsync(asum, 0, (size_t)N_NODES * 8 * sizeof(float), stream);
  hipMemsetAsync(keys,     0xFF, sizeof(unsigned), stream);  // min key = UINT_MAX
  hipMemsetAsync(keys + 1, 0x00, sizeof(unsigned), stream);  // max key = 0

  ts_minmax<<<1024, 256, 0, stream>>>(ts, keys, N_EDGES);
  qkv_gemm_wmma<<<N_NODES / 16, 256, 0, stream>>>(x, Wq, Wk, Wv, q, k, v);

  const int edge_blocks = (N_EDGES * 8) / 256;  // 50000, exact
  edge_pass1<<<edge_blocks, 256, 0, stream>>>(ei, eattr, ts, q, k, freq, tb, ew, eb,
                                              keys, aexp, asum);
  edge_pass2<<<edge_blocks, 256, 0, stream>>>(ei, v, aexp, asum, out);
}